// NMS_35914516529905
// MI455X (gfx1250) — compile-verified
//
#include <hip/hip_runtime.h>
#include <hip/hip_bf16.h>
#include <math.h>

// ---------------- problem constants ----------------
#define B_      8
#define P_      24564
#define ROW_    93        // 4 loc + 81 conf + 4 prior + 4 var
#define NFG_    80        // foreground classes (conf[...,1:])
#define TOPK_   400
#define KEEP_   200
#define GRID_   19
#define OUTC_   81
#define CONF_T  0.01f
#define NMS_T   0.45f
#define MASK_T  0.6f
#define NDET_   (NFG_ * TOPK_)          // 32000 per batch
#define OUTPB_  (GRID_ * GRID_ * OUTC_) // 29241 per batch
#define QCAP_   512

// ---------------- workspace layout (floats) ----------------
#define WS_BOXES     ((size_t)0)
#define WS_CSCORE    ((size_t)(B_ * P_ * 4))
#define WS_CBOX      (WS_CSCORE + (size_t)B_ * NDET_)
#define WS_QN        (WS_CBOX + (size_t)B_ * NDET_ * 4)
#define WS_QS        (WS_QN + B_)
#define WS_QCELL     (WS_QS + (size_t)B_ * QCAP_)
#define WS_QCH       (WS_QCELL + (size_t)B_ * QCAP_ * 4)

typedef __attribute__((ext_vector_type(4))) unsigned u32x4;
typedef __attribute__((ext_vector_type(8))) unsigned u32x8;

// ---------------- CDNA5 async global->LDS (per-lane) ----------------
// the VDST VGPR supplies the LDS destination address; generic shared pointers carry the LDS byte
// offset in their low 32 bits, so truncation yields the wave-relative address.
__device__ __forceinline__ void async_gather_b32(const void* gaddr, const void* lds_dst) {
  unsigned lds_off = (unsigned)(unsigned long long)lds_dst;
  unsigned long long ga = (unsigned long long)gaddr;
  asm volatile("global_load_async_to_lds_b32 %0, %1, off"
               :: "v"(lds_off), "v"(ga) : "memory");
}
__device__ __forceinline__ void async_gather_b128(const void* gaddr, const void* lds_dst) {
  unsigned lds_off = (unsigned)(unsigned long long)lds_dst;
  unsigned long long ga = (unsigned long long)gaddr;
  asm volatile("global_load_async_to_lds_b128 %0, %1, off"
               :: "v"(lds_off), "v"(ga) : "memory");
}
__device__ __forceinline__ void wait_async0() {
  asm volatile("s_wait_asynccnt 0" ::: "memory");
}

// ---------------- CDNA5 Tensor Data Mover: strided column -> LDS ----------
// One DMA descriptor gathers a 2-D tile (tile_dim0 elements per row,
// tile_dim1 rows, row stride = stride0 elements of 4B) into contiguous LDS.
// D# layout per CDNA5 ISA; VADDR2/3 = NULL (2-D).
__device__ __forceinline__ void tdm_load_tile_f32(const float* gptr, const void* lds_dst,
                                                  unsigned tile_d0, unsigned tile_d1,
                                                  unsigned stride0_elems) {
  unsigned long long ga = (unsigned long long)gptr;
  u32x4 g0;
  g0[0] = 1u;                                        // count=1, user descriptor
  g0[1] = (unsigned)(unsigned long long)lds_dst;     // lds_addr (bytes)
  g0[2] = (unsigned)ga;                              // global_addr[31:0]
  g0[3] = (unsigned)((ga >> 32) & 0x01FFFFFFull)     // global_addr[56:32]
          | (2u << 30);                              // type = 2 ("image")
  u32x8 g1;
  g1[0] = (2u << 16);                                // data_size = 4B; mask/flags = 0
  g1[1] = (tile_d0 & 0xFFFFu) << 16;                 // tensor_dim0[15:0] = tile_d0
  g1[2] = ((tile_d0 >> 16) & 0xFFFFu)                // tensor_dim0[31:16]
          | ((tile_d1 & 0xFFFFu) << 16);             // tensor_dim1[15:0]
  g1[3] = ((tile_d1 >> 16) & 0xFFFFu)                // tensor_dim1[31:16]
          | ((tile_d0 & 0xFFFFu) << 16);             // tile_dim0
  g1[4] = (tile_d1 & 0xFFFFu);                       // tile_dim1 (tile_dim2 = 0)
  g1[5] = stride0_elems;                             // tensor_dim0_stride[31:0]
  g1[6] = 0u;                                        // stride0[47:32], stride1[15:0]
  g1[7] = 0u;                                        // stride1[47:16]
  asm volatile("tensor_load_to_lds %0, %1" :: "s"(g0), "s"(g1) : "memory");
}

// ---------------- block reduction (wave32-aware) ----------------
__device__ __forceinline__ int blockSumBroadcast(int v, int* red /* >=9 ints */) {
  int lane = threadIdx.x & 31;
  int wid  = threadIdx.x >> 5;
  #pragma unroll
  for (int o = 16; o > 0; o >>= 1) v += __shfl_down(v, o);
  if (lane == 0) red[wid] = v;
  __syncthreads();
  if (threadIdx.x == 0) {
    int t = 0;
    int nw = (blockDim.x + 31) >> 5;
    for (int i = 0; i < nw; ++i) t += red[i];
    red[8] = t;
  }
  __syncthreads();
  return red[8];
}

// ======================================================================
// Kernel 1: decode boxes (replicates the reference, incl. cy using pw).
// ======================================================================
__global__ void k_decode(const float* __restrict__ x, float* __restrict__ boxes) {
  int t = blockIdx.x * blockDim.x + threadIdx.x;
  if (t >= B_ * P_) return;
  const float* r = x + (size_t)t * ROW_;
  float l0 = r[0], l1 = r[1], l2 = r[2], l3 = r[3];
  float p0 = r[85], p1 = r[86], p2 = r[87], p3 = r[88];
  float v0 = r[89], v1 = r[90], v2 = r[91], v3 = r[92];
  float pw = p2 - p0, ph = p3 - p1;
  float pcx = 0.5f * (p0 + p2), pcy = 0.5f * (p1 + p3);
  float cx = l0 * pw * v0 + pcx;
  float cy = l1 * pw * v1 + pcy;      // reference uses pw here, not ph
  float w  = expf(l2 * v2) * pw;
  float h  = expf(l3 * v3) * ph;
  float b0 = fminf(fmaxf(cx - 0.5f * w, 0.0f), 1.0f);
  float b1 = fminf(fmaxf(cy - 0.5f * h, 0.0f), 1.0f);
  float b2 = fminf(fmaxf(cx + 0.5f * w, 0.0f), 1.0f);
  float b3 = fminf(fmaxf(cy + 0.5f * h, 0.0f), 1.0f);
  float* o = boxes + (size_t)t * 4;
  o[0] = b0; o[1] = b1; o[2] = b2; o[3] = b3;
}

// ======================================================================
// Kernel 2: per-(batch,class) exact top-400. One block per (b,c).
// A single Tensor Data Mover descriptor DMA-gathers the strided score
// column (tile 1x24564, stride 93 elems) into a 96KB LDS tile (gfx1250:
// 320KB/WGP) while the waves idle on TENSORcnt. Then bit-level binary
// search for the exact 400th value, compaction, bitonic sort of 512.
// ======================================================================
__global__ void k_topk(const float* __restrict__ x,
                       const float* __restrict__ boxes,
                       float* __restrict__ candScore,
                       float* __restrict__ candBox) {
  extern __shared__ float smem[];          // P_ floats of this class's scores
  __shared__ float sKey[512];
  __shared__ int   sIdx[512];
  __shared__ int   red[9];
  __shared__ int   sCnt;
  __shared__ int   bIdx[64];
  __shared__ int   bCnt;

  const int blk = blockIdx.x;              // 0..639
  const int b   = blk / NFG_;
  const int c   = blk % NFG_;
  const int tid = threadIdx.x;

  // ---- TDM: DMA the strided score column into LDS ----
  if (tid < 32) {                          // wave 0 issues + waits (EXEC all-1s)
    const float* sp = x + (size_t)b * P_ * ROW_ + 5 + c;   // conf[...,1+c]
    tdm_load_tile_f32(sp, smem, /*tile_d0=*/1u, /*tile_d1=*/(unsigned)P_,
                      /*stride0=*/(unsigned)ROW_);
    __builtin_amdgcn_s_wait_tensorcnt(0);
  }
  __syncthreads();

  // ---- count above CONF_THRESH ----
  int lc = 0;
  for (int e = tid; e < P_; e += blockDim.x) lc += (smem[e] > CONF_T);
  int cnt01 = blockSumBroadcast(lc, red);

  unsigned loB = __float_as_uint(CONF_T);
  unsigned hiB = __float_as_uint(1.0f);
  if (cnt01 >= TOPK_) {
    for (int it = 0; it < 28; ++it) {      // bit-level binary search
      unsigned midB = (loB + hiB) >> 1;
      float midF = __uint_as_float(midB);
      int v = 0;
      for (int e = tid; e < P_; e += blockDim.x) v += (smem[e] > midF);
      int cnt = blockSumBroadcast(v, red);
      if (cnt >= TOPK_) loB = midB; else hiB = midB;
    }
  } else {
    hiB = loB;   // everything above CONF_T is selected
  }
  const float loF = __uint_as_float(loB);
  const float hiF = __uint_as_float(hiB);

  if (tid == 0) { sCnt = 0; bCnt = 0; }
  __syncthreads();

  // ---- definite selection: score > hiF; boundary values in (loF, hiF] ----
  for (int e = tid; e < P_; e += blockDim.x) {
    float s = smem[e];
    if (s > hiF) {
      int pos = atomicAdd(&sCnt, 1);
      if (pos < TOPK_) { sKey[pos] = s; sIdx[pos] = e; }
    } else if (s > loF) {
      int pos = atomicAdd(&bCnt, 1);
      if (pos < 64) bIdx[pos] = e;
    }
  }
  __syncthreads();

  // ---- fill from boundary, lowest index first (top_k tie-break) ----
  if (tid == 0) {
    int nb = bCnt < 64 ? bCnt : 64;
    for (int i = 1; i < nb; ++i) {         // insertion sort by index asc
      int key = bIdx[i], j = i - 1;
      while (j >= 0 && bIdx[j] > key) { bIdx[j + 1] = bIdx[j]; --j; }
      bIdx[j + 1] = key;
    }
    for (int i = 0; i < nb && sCnt < TOPK_; ++i) {
      sKey[sCnt] = smem[bIdx[i]];
      sIdx[sCnt] = bIdx[i];
      ++sCnt;
    }
  }
  __syncthreads();

  int nsel = sCnt;
  for (int i = tid; i < 512; i += blockDim.x) {
    if (i >= nsel) { sKey[i] = -1e30f; sIdx[i] = -1; }
  }
  __syncthreads();

  // ---- bitonic sort 512, descending by score, index asc on ties ----
  for (int k = 2; k <= 512; k <<= 1) {
    for (int j = k >> 1; j > 0; j >>= 1) {
      for (int i = tid; i < 512; i += blockDim.x) {
        int ixj = i ^ j;
        if (ixj > i) {
          float a = sKey[i], bb = sKey[ixj];
          int ia = sIdx[i], ib = sIdx[ixj];
          bool aFirst = (a > bb) || (a == bb && ia < ib);
          bool descSeg = ((i & k) == 0);
          if (descSeg ? !aFirst : aFirst) {
            sKey[i] = bb; sKey[ixj] = a;
            sIdx[i] = ib; sIdx[ixj] = ia;
          }
        }
      }
      __syncthreads();
    }
  }

  // ---- emit sorted candidates + gathered boxes ----
  for (int k = tid; k < TOPK_; k += blockDim.x) {
    size_t o = (size_t)blk * TOPK_ + k;
    int p = sIdx[k];
    if (p >= 0) {
      candScore[o] = sKey[k];
      const float* bb = boxes + ((size_t)b * P_ + p) * 4;
      candBox[o * 4 + 0] = bb[0];
      candBox[o * 4 + 1] = bb[1];
      candBox[o * 4 + 2] = bb[2];
      candBox[o * 4 + 3] = bb[3];
    } else {
      candScore[o] = -1.0f;
      candBox[o * 4 + 0] = 0.f; candBox[o * 4 + 1] = 0.f;
      candBox[o * 4 + 2] = 0.f; candBox[o * 4 + 3] = 0.f;
    }
  }
}

// ======================================================================
// Kernel 3: greedy NMS — one wave32 per (b,c). Each lane owns 13 slots
// entirely in VGPRs; box_i broadcast via __shfl, suppression via __any.
// No LDS, no barriers in the 400-iteration scan. Divide-free IoU test.
// ======================================================================
__global__ void __launch_bounds__(32)
k_nms(float* __restrict__ candScore, const float* __restrict__ candBox) {
  const int blk  = blockIdx.x;       // (b,c)
  const int lane = threadIdx.x;      // 0..31

  float x1[13], y1[13], x2[13], y2[13], ar[13], scr[13];
  unsigned validMask = 0, keptMask = 0;

  #pragma unroll
  for (int r = 0; r < 13; ++r) {
    int s = r * 32 + lane;
    bool in = s < TOPK_;
    size_t o = (size_t)blk * TOPK_ + (in ? s : 0);
    float sc = in ? candScore[o] : -1.0f;
    float a0 = in ? candBox[o * 4 + 0] : 0.f;
    float a1 = in ? candBox[o * 4 + 1] : 0.f;
    float a2 = in ? candBox[o * 4 + 2] : 0.f;
    float a3 = in ? candBox[o * 4 + 3] : 0.f;
    x1[r] = a0; y1[r] = a1; x2[r] = a2; y2[r] = a3;
    ar[r] = (a2 - a0) * (a3 - a1);
    scr[r] = sc;
    if (in && sc > CONF_T) validMask |= (1u << r);
  }

  #pragma unroll
  for (int ri = 0; ri < 13; ++ri) {
    const int nlane = (ri == 12) ? 16 : 32;   // slots 0..399
    for (int li = 0; li < nlane; ++li) {
      float ix1 = __shfl(x1[ri], li);
      float iy1 = __shfl(y1[ri], li);
      float ix2 = __shfl(x2[ri], li);
      float iy2 = __shfl(y2[ri], li);
      float iar = __shfl(ar[ri], li);
      bool sup = false;
      #pragma unroll
      for (int r = 0; r < 13; ++r) {
        if (keptMask & (1u << r)) {
          float xx1 = fmaxf(x1[r], ix1);
          float yy1 = fmaxf(y1[r], iy1);
          float xx2 = fminf(x2[r], ix2);
          float yy2 = fminf(y2[r], iy2);
          float inter = fmaxf(xx2 - xx1, 0.f) * fmaxf(yy2 - yy1, 0.f);
          float uni = ar[r] + iar - inter;
          sup |= (uni > 0.f) & (inter > NMS_T * uni);   // iou > T, no divide
        }
      }
      bool any = __any(sup);
      if (lane == li && (validMask & (1u << ri)) && !any)
        keptMask |= (1u << ri);
    }
  }

  #pragma unroll
  for (int r = 0; r < 13; ++r) {
    int s = r * 32 + lane;
    if (s < TOPK_) {
      size_t o = (size_t)blk * TOPK_ + s;
      candScore[o] = (keptMask & (1u << r)) ? scr[r] : -1.0f;
    }
  }
}

// ======================================================================
// Kernel 4: per-batch KEEP_TOP_K cutoff + compaction of qualified
// detections (score >= max(cutoff, 0.6)). Rank order is not needed:
// the reference's win=max(k) == min score among qualified coverers.
// ======================================================================
__global__ void k_cutoff(const float* __restrict__ candScore,
                         const float* __restrict__ candBox,
                         int* __restrict__ qn, float* __restrict__ qs,
                         int* __restrict__ qcell, int* __restrict__ qch) {
  __shared__ int red[9];
  __shared__ int cnt;
  __shared__ float ls[QCAP_];
  __shared__ int   lcell[QCAP_ * 4];
  __shared__ int   lch[QCAP_];

  const int b = blockIdx.x;
  const int tid = threadIdx.x;
  const float* base = candScore + (size_t)b * NDET_;

  unsigned loB = 0u;                       // 0.0f
  unsigned hiB = __float_as_uint(1.0f);
  for (int it = 0; it < 30; ++it) {
    unsigned midB = (loB + hiB) >> 1;
    float midF = __uint_as_float(midB);
    int v = 0;
    for (int e = tid; e < NDET_; e += blockDim.x) v += (base[e] > midF);
    int c = blockSumBroadcast(v, red);
    if (c >= KEEP_) loB = midB; else hiB = midB;
  }
  const float loF = __uint_as_float(loB);

  if (tid == 0) cnt = 0;
  __syncthreads();

  for (int e = tid; e < NDET_; e += blockDim.x) {
    float s = base[e];
    if (s >= MASK_T && s > loF) {
      int pos = atomicAdd(&cnt, 1);
      if (pos < QCAP_) {
        ls[pos] = s;
        const float* bb = candBox + ((size_t)b * NDET_ + e) * 4;
        lcell[pos * 4 + 0] = (int)rintf(bb[0] * (float)GRID_); // round-half-even
        lcell[pos * 4 + 1] = (int)rintf(bb[1] * (float)GRID_);
        lcell[pos * 4 + 2] = (int)rintf(bb[2] * (float)GRID_);
        lcell[pos * 4 + 3] = (int)rintf(bb[3] * (float)GRID_);
        lch[pos] = e / TOPK_;              // class channel 0..79
      }
    }
  }
  __syncthreads();

  int n = cnt < QCAP_ ? cnt : QCAP_;
  if (tid == 0) qn[b] = n;
  for (int j = tid; j < n; j += blockDim.x) {
    qs[b * QCAP_ + j] = ls[j];
    qch[b * QCAP_ + j] = lch[j];
    qcell[(b * QCAP_ + j) * 4 + 0] = lcell[j * 4 + 0];
    qcell[(b * QCAP_ + j) * 4 + 1] = lcell[j * 4 + 1];
    qcell[(b * QCAP_ + j) * 4 + 2] = lcell[j * 4 + 2];
    qcell[(b * QCAP_ + j) * 4 + 3] = lcell[j * 4 + 3];
  }
}

// ======================================================================
// Kernel 5: rasterize. Per-batch qualified list is staged into LDS with
// CDNA5 async global->LDS loads (overlapped, then s_wait_asynccnt).
// out[b,y,x,ch] = min score among qualified covering detections, else 0.
// ======================================================================
__global__ void k_raster(const int* __restrict__ qn, const float* __restrict__ qs,
                         const int* __restrict__ qcell, const int* __restrict__ qch,
                         float* __restrict__ out) {
  __shared__ float ss[QCAP_];
  __shared__ int scell[QCAP_ * 4];
  __shared__ int sch[QCAP_];

  const int b = blockIdx.y;
  const int tid = threadIdx.x;

  // async-stage the full capacity (entries beyond n are never read)
  for (int j = tid; j < QCAP_; j += blockDim.x) {
    async_gather_b32(&qs[b * QCAP_ + j], &ss[j]);
    async_gather_b32(&qch[b * QCAP_ + j], &sch[j]);
    async_gather_b128(&qcell[(b * QCAP_ + j) * 4], &scell[j * 4]);
  }
  const int n = qn[b];
  wait_async0();
  __syncthreads();

  int idx = blockIdx.x * blockDim.x + tid;
  if (idx >= OUTPB_) return;
  int ch = idx % OUTC_;
  int pos = idx / OUTC_;
  int gx = pos % GRID_;
  int gy = pos / GRID_;

  float best = 1e30f;
  bool found = false;
  for (int j = 0; j < n; ++j) {
    if (sch[j] == ch &&
        gy >= scell[j * 4 + 1] && gy < scell[j * 4 + 3] &&
        gx >= scell[j * 4 + 0] && gx < scell[j * 4 + 2]) {
      float s = ss[j];
      if (s < best) { best = s; found = true; }
    }
  }
  out[(size_t)b * OUTPB_ + idx] = found ? best : 0.0f;
}

// ======================================================================
extern "C" void kernel_launch(void* const* d_in, const int* in_sizes, int n_in,
                              void* d_out, int out_size, void* d_ws, size_t ws_size,
                              hipStream_t stream) {
  (void)in_sizes; (void)n_in; (void)out_size; (void)ws_size;
  const float* x = (const float*)d_in[0];
  float* out = (float*)d_out;
  float* ws = (float*)d_ws;

  float* boxes     = ws + WS_BOXES;
  float* candScore = ws + WS_CSCORE;
  float* candBox   = ws + WS_CBOX;
  int*   qn        = (int*)(ws + WS_QN);
  float* qs        = ws + WS_QS;
  int*   qcell     = (int*)(ws + WS_QCELL);
  int*   qch       = (int*)(ws + WS_QCH);

  // 1. decode all boxes (streaming pass; x ~73MB fits in 192MB L2 afterwards)
  k_decode<<<(B_ * P_ + 255) / 256, 256, 0, stream>>>(x, boxes);

  // 2. per-(b,c) exact top-400 — TDM DMA of the strided score column into
  //    a 96KB dynamic-LDS tile (gfx1250: 320KB/WGP)
  k_topk<<<B_ * NFG_, 256, (size_t)P_ * sizeof(float), stream>>>(
      x, boxes, candScore, candBox);

  // 3. wave32 register-resident greedy NMS, one wave per (b,c)
  k_nms<<<B_ * NFG_, 32, 0, stream>>>(candScore, candBox);

  // 4. per-batch top-200 cutoff + qualified compaction
  k_cutoff<<<B_, 256, 0, stream>>>(candScore, candBox, qn, qs, qcell, qch);

  // 5. rasterize 19x19x81 per batch (async-staged detection list)
  k_raster<<<dim3((OUTPB_ + 255) / 256, B_), 256, 0, stream>>>(
      qn, qs, qcell, qch, out);
}